// ParametricKernelLayer_68247030334082
// MI455X (gfx1250) — compile-verified
//
#include <hip/hip_runtime.h>
#include <hip/hip_bf16.h>

#define NF 8
#define NK 16
#define BB 32
#define YY 32
#define XX 32
#define PPP 16
#define TTT 8
#define YXPT (YY*XX*PPP*TTT)     // 131072
#define NCHUNK 256
#define CHUNK (YXPT/NCHUNK)      // 512 = 4 xy-blocks of 128 (= P*T)

// workspace offsets (in floats)
#define QMEAN_OFF 0
#define TAB_OFF   131072
#define KY_OFF    (TAB_OFF)            // NF*NK*YY  = 4096
#define KX_OFF    (TAB_OFF + 4096)    // NF*NK*XX  = 4096
#define KP_OFF    (TAB_OFF + 8192)    // NF*NK*PPP = 2048
#define KT_OFF    (TAB_OFF + 10240)   // NF*NK*TTT = 1024
#define INV_OFF   (TAB_OFF + 11264)   // 128
#define PART_OFF  (INV_OFF + 128)     // NCHUNK*BB*NF*NK = 1048576

typedef __attribute__((ext_vector_type(2))) float v2f;
typedef __attribute__((ext_vector_type(8))) float v8f;

// ---- 1: build separable 1-D kernel tables (f,k) x {y,x,p,t} ----
__global__ void pk_tables(const float* __restrict__ mean_lat, const float* __restrict__ logstd_lat,
                          const float* __restrict__ mean_lon, const float* __restrict__ logstd_lon,
                          const float* __restrict__ mean_lev, const float* __restrict__ logstd_lev,
                          const float* __restrict__ logtau,   float* __restrict__ ws) {
    int fk = threadIdx.x;
    if (fk >= NF*NK) return;
    float mla = mean_lat[fk], sla = expf(logstd_lat[fk]);
    float mlo = mean_lon[fk], slo = expf(logstd_lon[fk]);
    float mle = mean_lev[fk], sle = expf(logstd_lev[fk]);
    float tau = expf(logtau[fk]) + 1e-4f;
    for (int y = 0; y < YY; ++y) {
        float c = -1.f + 2.f * (float)y / (float)(YY - 1);
        float z = (c - mla) / sla;
        ws[KY_OFF + fk*YY + y] = expf(-0.5f * z * z);
    }
    for (int x = 0; x < XX; ++x) {
        float c = -1.f + 2.f * (float)x / (float)(XX - 1);
        float z = (c - mlo) / slo;
        ws[KX_OFF + fk*XX + x] = expf(-0.5f * z * z);
    }
    for (int p = 0; p < PPP; ++p) {
        float c = -1.f + 2.f * (float)p / (float)(PPP - 1);
        float z = (c - mle) / sle;
        ws[KP_OFF + fk*PPP + p] = expf(-0.5f * z * z);
    }
    for (int t = 0; t < TTT; ++t)
        ws[KT_OFF + fk*TTT + t] = expf(-(float)t / tau);
}

// ---- 2: qmean over batch ----
__global__ void pk_qmean(const float* __restrict__ quad, float* __restrict__ ws) {
    int i = blockIdx.x * blockDim.x + threadIdx.x;
    if (i >= YXPT) return;
    float acc = 0.f;
    for (int b = 0; b < BB; ++b) acc += quad[(size_t)b * YXPT + i];
    ws[QMEAN_OFF + i] = acc * (1.0f / (float)BB);
}

// ---- 3: integral[f,k] = sum w*qmean ; store reciprocal ----
__global__ void pk_integral(float* __restrict__ ws) {
    __shared__ float sk[88];
    __shared__ float red[256];
    const int fk = blockIdx.x, tid = threadIdx.x;
    if (tid < 32)       sk[tid] = ws[KY_OFF + fk*YY  + tid];
    else if (tid < 64)  sk[tid] = ws[KX_OFF + fk*XX  + (tid - 32)];
    else if (tid < 80)  sk[tid] = ws[KP_OFF + fk*PPP + (tid - 64)];
    else if (tid < 88)  sk[tid] = ws[KT_OFF + fk*TTT + (tid - 80)];
    __syncthreads();
    float acc = 0.f;
    for (int i = tid; i < YXPT; i += 256) {
        int t = i & 7, p = (i >> 3) & 15, x = (i >> 7) & 31, y = (i >> 12) & 31;
        float w = sk[y] * sk[32 + x] * sk[64 + p] * sk[80 + t];
        acc += w * ws[QMEAN_OFF + i];
    }
    red[tid] = acc;
    __syncthreads();
    for (int s = 128; s > 0; s >>= 1) {
        if (tid < s) red[tid] += red[tid + s];
        __syncthreads();
    }
    if (tid == 0) ws[INV_OFF + fk] = 1.0f / (red[0] + 1e-4f);
}

// ---- 4: main split-K GEMM with fp32 WMMA ----
// grid = (NCHUNK, NF), block = 32 (one wave32; EXEC all ones for WMMA).
// Loop order: xy-block (hoists ky*kx per 128 elems) -> p (hoists *kp per 8)
// -> 2 K-steps over t (kt values held in registers).
__global__ __launch_bounds__(32) void pk_wmma(const float* __restrict__ field,
                                              const float* __restrict__ quad,
                                              const float* __restrict__ ws,
                                              float* __restrict__ partials) {
    __shared__ float sk[1408];   // ky[16][32] | kx[16][32] | kp[16][16] | kt[16][8], this f only
    const int chunk = blockIdx.x;
    const int f     = blockIdx.y;
    const int lane  = threadIdx.x;
    for (int i = lane; i < 512; i += 32) sk[i]        = ws[KY_OFF + f*512 + i];
    for (int i = lane; i < 512; i += 32) sk[512 + i]  = ws[KX_OFF + f*512 + i];
    for (int i = lane; i < 256; i += 32) sk[1024 + i] = ws[KP_OFF + f*256 + i];
    for (int i = lane; i < 128; i += 32) sk[1280 + i] = ws[KT_OFF + f*128 + i];
    __syncthreads();

    const int n  = lane & 15;          // A row (b within tile) and B column (k)
    const int kb = (lane >> 4) << 1;   // K-offset pair: lanes 0-15 -> {0,1}, 16-31 -> {2,3}

    const float* f0 = field + ((size_t)n        * NF + f) * YXPT;   // b = n
    const float* f1 = field + ((size_t)(n + 16) * NF + f) * YXPT;   // b = n+16
    const float* q0 = quad  + (size_t)n        * YXPT;
    const float* q1 = quad  + (size_t)(n + 16) * YXPT;

    // this lane's 4 kt values (t = kb, kb+1, 4+kb, 4+kb+1) kept in registers
    const float ktA0 = sk[1280 + n*8 + kb];
    const float ktA1 = sk[1280 + n*8 + kb + 1];
    const float ktB0 = sk[1280 + n*8 + 4 + kb];
    const float ktB1 = sk[1280 + n*8 + 4 + kb + 1];

    v8f c0 = {};
    v8f c1 = {};
    const int base = chunk * CHUNK;
    for (int j = 0; j < CHUNK / (PPP * TTT); ++j) {       // 4 xy-blocks of 128
        const int xyi = (base >> 7) + j;
        const int y = xyi >> 5, x = xyi & 31;
        const float kyx = sk[n*32 + y] * sk[512 + n*32 + x];
        const int ib = base + j * (PPP * TTT);
        // stream prefetch one xy-block ahead (speculative; OOB silently dropped)
        __builtin_prefetch(f0 + ib + 128, 0, 1);
        __builtin_prefetch(f1 + ib + 128, 0, 1);
        __builtin_prefetch(q0 + ib + 128, 0, 1);
        __builtin_prefetch(q1 + ib + 128, 0, 1);
        for (int p = 0; p < PPP; ++p) {
            const float common = kyx * sk[1024 + n*16 + p];
            const int i8 = ib + p * TTT;
            // K-step A: t = kb, kb+1
            {
                const int i0 = i8 + kb;
                v2f a0, a1, bm;
                a0.x = f0[i0] * q0[i0];  a0.y = f0[i0+1] * q0[i0+1];
                a1.x = f1[i0] * q1[i0];  a1.y = f1[i0+1] * q1[i0+1];
                bm.x = common * ktA0;    bm.y = common * ktA1;
                c0 = __builtin_amdgcn_wmma_f32_16x16x4_f32(false, a0, false, bm, (short)0, c0, false, false);
                c1 = __builtin_amdgcn_wmma_f32_16x16x4_f32(false, a1, false, bm, (short)0, c1, false, false);
            }
            // K-step B: t = 4+kb, 4+kb+1
            {
                const int i0 = i8 + 4 + kb;
                v2f a0, a1, bm;
                a0.x = f0[i0] * q0[i0];  a0.y = f0[i0+1] * q0[i0+1];
                a1.x = f1[i0] * q1[i0];  a1.y = f1[i0+1] * q1[i0+1];
                bm.x = common * ktB0;    bm.y = common * ktB1;
                c0 = __builtin_amdgcn_wmma_f32_16x16x4_f32(false, a0, false, bm, (short)0, c0, false, false);
                c1 = __builtin_amdgcn_wmma_f32_16x16x4_f32(false, a1, false, bm, (short)0, c1, false, false);
            }
        }
    }

    // C/D layout: lane holds N = lane&15; VGPR r holds M = r + 8*(lane>>4)
    const int half = lane >> 4;
    float* pb = partials + (size_t)chunk * (BB * NF * NK);
#pragma unroll
    for (int r = 0; r < 8; ++r) {
        int b0 = r + 8 * half;
        pb[((size_t)b0        * NF + f) * NK + n] = c0[r];
        pb[((size_t)(b0 + 16) * NF + f) * NK + n] = c1[r];
    }
}

// ---- 5: reduce partials deterministically, scale by 1/integral ----
__global__ void pk_finalize(const float* __restrict__ ws, float* __restrict__ out) {
    int idx = blockIdx.x * blockDim.x + threadIdx.x;   // 0 .. B*NF*NK-1 (4096)
    if (idx >= BB * NF * NK) return;
    int fk = idx & (NF * NK - 1);
    float acc = 0.f;
    const float* p = ws + PART_OFF + idx;
    for (int c = 0; c < NCHUNK; ++c) acc += p[(size_t)c * (BB * NF * NK)];
    out[idx] = acc * ws[INV_OFF + fk];
}

extern "C" void kernel_launch(void* const* d_in, const int* in_sizes, int n_in,
                              void* d_out, int out_size, void* d_ws, size_t ws_size,
                              hipStream_t stream) {
    const float* field      = (const float*)d_in[0];
    const float* quad       = (const float*)d_in[1];
    const float* mean_lat   = (const float*)d_in[2];
    const float* logstd_lat = (const float*)d_in[3];
    const float* mean_lon   = (const float*)d_in[4];
    const float* logstd_lon = (const float*)d_in[5];
    const float* mean_lev   = (const float*)d_in[6];
    const float* logstd_lev = (const float*)d_in[7];
    const float* logtau     = (const float*)d_in[8];
    float* ws  = (float*)d_ws;
    float* out = (float*)d_out;

    pk_tables  <<<1, 128, 0, stream>>>(mean_lat, logstd_lat, mean_lon, logstd_lon,
                                       mean_lev, logstd_lev, logtau, ws);
    pk_qmean   <<<YXPT / 256, 256, 0, stream>>>(quad, ws);
    pk_integral<<<NF * NK, 256, 0, stream>>>(ws);
    pk_wmma    <<<dim3(NCHUNK, NF), 32, 0, stream>>>(field, quad, ws, ws + PART_OFF);
    pk_finalize<<<(BB * NF * NK) / 256, 256, 0, stream>>>(ws, out);
}